// CustomModule_63024350102067
// MI455X (gfx1250) — compile-verified
//
#include <hip/hip_runtime.h>
#include <math.h>

// Problem constants (fixed by the reference's setup_inputs)
#define Bn  16
#define Cn  512
#define HWn 4096
#define CRn 32

typedef __attribute__((ext_vector_type(2))) float v2f;
typedef __attribute__((ext_vector_type(8))) float v8f;

// ---------------------------------------------------------------------------
// Kernel 1/4: row reduction over HW.
//   out[row] = scale * sum_n x[row, n] * (w ? w[b, n] : 1)
// row = b*C + c. One 256-thread block per row, float4 coalesced loads,
// deterministic LDS tree reduction.
// ---------------------------------------------------------------------------
__global__ void row_reduce_kernel(const float* __restrict__ x,
                                  const float* __restrict__ w,   // [B*HW] or nullptr
                                  float* __restrict__ out, float scale) {
  const int row = blockIdx.x;          // b*C + c
  const int b   = row >> 9;            // /C (C=512)
  const int tid = threadIdx.x;
  const float4* x4 = (const float4*)(x + (size_t)row * HWn);
  float acc = 0.0f;
  if (w != nullptr) {
    const float4* w4 = (const float4*)(w + (size_t)b * HWn);
#pragma unroll
    for (int j = 0; j < 4; ++j) {
      const int i = j * 256 + tid;
      const float4 xv = x4[i];
      const float4 wv = w4[i];
      acc += xv.x * wv.x + xv.y * wv.y + xv.z * wv.z + xv.w * wv.w;
    }
  } else {
#pragma unroll
    for (int j = 0; j < 4; ++j) {
      const int i = j * 256 + tid;
      const float4 xv = x4[i];
      acc += xv.x + xv.y + xv.z + xv.w;
    }
  }
  __shared__ float red[256];
  red[tid] = acc;
  __syncthreads();
  for (int s = 128; s > 0; s >>= 1) {
    if (tid < s) red[tid] += red[tid + s];
    __syncthreads();
  }
  if (tid == 0) out[row] = red[0] * scale;
}

// ---------------------------------------------------------------------------
// Kernel 2/5: column dot over C.
//   out[b, n] = (maybe sigmoid)( bias + sum_c x[b, c, n] * vec[b*stride + c] )
// grid: (HW/1024, B); each thread owns 4 consecutive n (float4).
// Coalesced b128 loads per channel row; prefetch hints ahead 8 channels.
// ---------------------------------------------------------------------------
__global__ void col_dot_kernel(const float* __restrict__ x,
                               const float* __restrict__ vec, int vecStrideB,
                               const float* __restrict__ bias,
                               float* __restrict__ out, int applySigmoid) {
  const int b   = blockIdx.y;
  const int tid = threadIdx.x;
  const int n4  = blockIdx.x * 256 + tid;        // float4 index in [0, HW/4)
  const float* xb = x + (size_t)b * Cn * HWn;
  const float* v  = vec + (size_t)b * vecStrideB;

  float ax = 0.f, ay = 0.f, az = 0.f, aw = 0.f;
  if (bias != nullptr) { const float bv = bias[0]; ax = ay = az = aw = bv; }

  for (int c = 0; c < Cn; c += 4) {
    // speculative prefetch of upcoming channel rows -> global_prefetch_b8
    __builtin_prefetch(xb + (size_t)(c + 8) * HWn + (size_t)n4 * 4, 0, 0);
#pragma unroll
    for (int u = 0; u < 4; ++u) {
      const float  wv = v[c + u];
      const float4 xv = ((const float4*)(xb + (size_t)(c + u) * HWn))[n4];
      ax = fmaf(wv, xv.x, ax);
      ay = fmaf(wv, xv.y, ay);
      az = fmaf(wv, xv.z, az);
      aw = fmaf(wv, xv.w, aw);
    }
  }
  if (applySigmoid) {
    ax = 1.0f / (1.0f + __expf(-ax));
    ay = 1.0f / (1.0f + __expf(-ay));
    az = 1.0f / (1.0f + __expf(-az));
    aw = 1.0f / (1.0f + __expf(-aw));
  }
  float4 o; o.x = ax; o.y = ay; o.z = az; o.w = aw;
  ((float4*)(out + (size_t)b * HWn))[n4] = o;
}

// ---------------------------------------------------------------------------
// Kernel 3: softmax over HW per batch. One block per b; 16 values per thread.
// ---------------------------------------------------------------------------
__global__ void softmax_kernel(const float* __restrict__ logits,
                               float* __restrict__ attn) {
  const int b   = blockIdx.x;
  const int tid = threadIdx.x;
  const float4* in4 = (const float4*)(logits + (size_t)b * HWn);
  float4 v[4];
  float mx = -INFINITY;
#pragma unroll
  for (int j = 0; j < 4; ++j) {
    v[j] = in4[j * 256 + tid];
    mx = fmaxf(mx, fmaxf(fmaxf(v[j].x, v[j].y), fmaxf(v[j].z, v[j].w)));
  }
  __shared__ float red[256];
  red[tid] = mx;
  __syncthreads();
  for (int s = 128; s > 0; s >>= 1) {
    if (tid < s) red[tid] = fmaxf(red[tid], red[tid + s]);
    __syncthreads();
  }
  mx = red[0];
  __syncthreads();

  float sum = 0.0f;
#pragma unroll
  for (int j = 0; j < 4; ++j) {
    v[j].x = __expf(v[j].x - mx);
    v[j].y = __expf(v[j].y - mx);
    v[j].z = __expf(v[j].z - mx);
    v[j].w = __expf(v[j].w - mx);
    sum += v[j].x + v[j].y + v[j].z + v[j].w;
  }
  red[tid] = sum;
  __syncthreads();
  for (int s = 128; s > 0; s >>= 1) {
    if (tid < s) red[tid] += red[tid + s];
    __syncthreads();
  }
  const float inv = 1.0f / red[0];

  float4* out4 = (float4*)(attn + (size_t)b * HWn);
#pragma unroll
  for (int j = 0; j < 4; ++j) {
    float4 o;
    o.x = v[j].x * inv; o.y = v[j].y * inv; o.z = v[j].z * inv; o.w = v[j].w * inv;
    out4[j * 256 + tid] = o;
  }
}

// ---------------------------------------------------------------------------
// Kernel 6: bottleneck MLP entirely via V_WMMA_F32_16X16X4_F32.
// Single wave (32 lanes), uniform control flow => EXEC all-ones for WMMA.
//   t1[16,32] = ctx[16,512] @ w2^T + b2          (2 N-tiles x 128 k-steps)
//   t2[16,32] = t1 @ w3^T + b3                   (2 N-tiles x 8 k-steps)
//   LayerNorm(rows of 32) * ln_g + ln_b, ReLU
//   m[16,512] = t2 @ w4^T + b4                   (32 N-tiles x 8 k-steps)
// Fragment layouts per CDNA5 ISA 7.12.2:
//   A (16x4, f32):  lane%16 = M;  v0/v1 hold K = (lane<16 ? 0,1 : 2,3)
//   B (4x16, f32):  lane%16 = N;  v0/v1 hold K = (lane<16 ? 0,1 : 2,3)
//   C/D (16x16):    lane%16 = N;  v[i] holds M = i + (lane<16 ? 0 : 8)
// ---------------------------------------------------------------------------
__global__ void mlp_wmma_kernel(const float* __restrict__ ctx,
                                const float* __restrict__ w2, const float* __restrict__ b2,
                                const float* __restrict__ w3, const float* __restrict__ b3,
                                const float* __restrict__ ln_g, const float* __restrict__ ln_b,
                                const float* __restrict__ w4, const float* __restrict__ b4,
                                float* __restrict__ m) {
  __shared__ float t1s[16 * 32];
  __shared__ float t2s[16 * 32];
  __shared__ float muS[16];
  __shared__ float rsS[16];

  const int lane  = threadIdx.x;        // 0..31
  const int nrow  = lane & 15;          // N col for B/C/D, M row for A
  const int khi   = (lane >> 4) << 1;   // 0 or 2
  const int mbase = (lane >> 4) << 3;   // 0 or 8

  // ---- t1 = ctx @ w2^T + b2 ------------------------------------------------
  for (int nt = 0; nt < 2; ++nt) {
    const int nb = nt * 16 + nrow;
    const float bv = b2[nb];
    v8f acc = {bv, bv, bv, bv, bv, bv, bv, bv};
#pragma unroll 4
    for (int k0 = 0; k0 < Cn; k0 += 4) {
      const int k = k0 + khi;
      v2f a  = {ctx[nrow * Cn + k], ctx[nrow * Cn + k + 1]};
      v2f bf = {w2[nb * Cn + k],    w2[nb * Cn + k + 1]};
      acc = __builtin_amdgcn_wmma_f32_16x16x4_f32(false, a, false, bf,
                                                  (short)0, acc, false, false);
    }
#pragma unroll
    for (int i = 0; i < 8; ++i) t1s[(mbase + i) * 32 + nt * 16 + nrow] = acc[i];
  }
  __syncthreads();

  // ---- t2 = t1 @ w3^T + b3 -------------------------------------------------
  for (int nt = 0; nt < 2; ++nt) {
    const int nb = nt * 16 + nrow;
    const float bv = b3[nb];
    v8f acc = {bv, bv, bv, bv, bv, bv, bv, bv};
#pragma unroll
    for (int k0 = 0; k0 < CRn; k0 += 4) {
      const int k = k0 + khi;
      v2f a  = {t1s[nrow * CRn + k], t1s[nrow * CRn + k + 1]};
      v2f bf = {w3[nb * CRn + k],    w3[nb * CRn + k + 1]};
      acc = __builtin_amdgcn_wmma_f32_16x16x4_f32(false, a, false, bf,
                                                  (short)0, acc, false, false);
    }
#pragma unroll
    for (int i = 0; i < 8; ++i) t2s[(mbase + i) * 32 + nt * 16 + nrow] = acc[i];
  }
  __syncthreads();

  // ---- LayerNorm (biased var over 32) + affine + ReLU ----------------------
  if (lane < 16) {
    float s = 0.f, s2 = 0.f;
    for (int r = 0; r < 32; ++r) {
      const float v = t2s[lane * 32 + r];
      s += v; s2 += v * v;
    }
    const float mu  = s * (1.0f / 32.0f);
    const float var = s2 * (1.0f / 32.0f) - mu * mu;
    muS[lane] = mu;
    rsS[lane] = rsqrtf(var + 1e-5f);
  }
  __syncthreads();
  for (int j = 0; j < 16; ++j) {
    const int idx = lane * 16 + j;       // 32 lanes * 16 = 512 values
    const int bb  = idx >> 5;
    const int r   = idx & 31;
    const float v = (t2s[idx] - muS[bb]) * rsS[bb] * ln_g[r] + ln_b[r];
    t2s[idx] = fmaxf(v, 0.0f);
  }
  __syncthreads();

  // ---- m = t2 @ w4^T + b4 --------------------------------------------------
  for (int nt = 0; nt < 32; ++nt) {
    const int nb = nt * 16 + nrow;       // output channel
    const float bv = b4[nb];
    v8f acc = {bv, bv, bv, bv, bv, bv, bv, bv};
#pragma unroll
    for (int k0 = 0; k0 < CRn; k0 += 4) {
      const int k = k0 + khi;
      v2f a  = {t2s[nrow * CRn + k], t2s[nrow * CRn + k + 1]};
      v2f bf = {w4[nb * CRn + k],    w4[nb * CRn + k + 1]};
      acc = __builtin_amdgcn_wmma_f32_16x16x4_f32(false, a, false, bf,
                                                  (short)0, acc, false, false);
    }
#pragma unroll
    for (int i = 0; i < 8; ++i) m[(mbase + i) * Cn + nb] = acc[i];
  }
}

// ---------------------------------------------------------------------------
// Kernel 7: out[b,c,n] = m[b,c] * sig[b,n] + x[b,c,n]
// ---------------------------------------------------------------------------
__global__ void final_kernel(const float* __restrict__ x,
                             const float* __restrict__ m,
                             const float* __restrict__ sig,
                             float* __restrict__ out) {
  const int row = blockIdx.y;            // b*C + c
  const int b   = row >> 9;
  const int tid = threadIdx.x;
  const int n4  = blockIdx.x * 256 + tid;
  const float mv = m[row];
  const float4 xv = ((const float4*)(x   + (size_t)row * HWn))[n4];
  const float4 sv = ((const float4*)(sig + (size_t)b   * HWn))[n4];
  float4 o;
  o.x = fmaf(mv, sv.x, xv.x);
  o.y = fmaf(mv, sv.y, xv.y);
  o.z = fmaf(mv, sv.z, xv.z);
  o.w = fmaf(mv, sv.w, xv.w);
  ((float4*)(out + (size_t)row * HWn))[n4] = o;
}

// ---------------------------------------------------------------------------
extern "C" void kernel_launch(void* const* d_in, const int* in_sizes, int n_in,
                              void* d_out, int out_size, void* d_ws, size_t ws_size,
                              hipStream_t stream) {
  const float* x    = (const float*)d_in[0];
  const float* w1   = (const float*)d_in[1];
  const float* b1   = (const float*)d_in[2];
  const float* w2   = (const float*)d_in[3];
  const float* b2   = (const float*)d_in[4];
  const float* w3   = (const float*)d_in[5];
  const float* b3   = (const float*)d_in[6];
  const float* ln_g = (const float*)d_in[7];
  const float* ln_b = (const float*)d_in[8];
  const float* w4   = (const float*)d_in[9];
  const float* b4   = (const float*)d_in[10];
  float* out = (float*)d_out;

  float* ws     = (float*)d_ws;
  float* gap    = ws;                    // B*C   = 8192
  float* logits = gap    + Bn * Cn;      // B*HW  = 65536
  float* attn   = logits + Bn * HWn;     // B*HW
  float* sig    = attn   + Bn * HWn;     // B*HW
  float* ctx    = sig    + Bn * HWn;     // B*C
  float* mbuf   = ctx    + Bn * Cn;      // B*C
  (void)in_sizes; (void)n_in; (void)out_size; (void)ws_size;

  // Pass 1: gap[b,c] = mean_n x
  row_reduce_kernel<<<Bn * Cn, 256, 0, stream>>>(x, nullptr, gap, 1.0f / (float)HWn);
  // Pass 2: logits[b,n] = sum_c x * w1[c] + b1
  col_dot_kernel<<<dim3(HWn / 1024, Bn), 256, 0, stream>>>(x, w1, 0, b1, logits, 0);
  // Softmax over HW
  softmax_kernel<<<Bn, 256, 0, stream>>>(logits, attn);
  // Pass 3: ctx[b,c] = sum_n x * attn[b,n]
  row_reduce_kernel<<<Bn * Cn, 256, 0, stream>>>(x, attn, ctx, 1.0f);
  // Pass 4: sig[b,n] = sigmoid( sum_c gap[b,c] * x )
  col_dot_kernel<<<dim3(HWn / 1024, Bn), 256, 0, stream>>>(x, gap, Cn, nullptr, sig, 1);
  // Bottleneck MLP via WMMA f32 16x16x4 (single wave)
  mlp_wmma_kernel<<<1, 32, 0, stream>>>(ctx, w2, b2, w3, b3, ln_g, ln_b, w4, b4, mbuf);
  // Pass 5: out = m * sig + x
  final_kernel<<<dim3(HWn / 1024, Bn * Cn), 256, 0, stream>>>(x, mbuf, sig, out);
}